// simple_GCN_83459804496108
// MI455X (gfx1250) — compile-verified
//
#include <hip/hip_runtime.h>

// ---------------------------------------------------------------------------
// 2-layer GCN for MI455X (gfx1250, wave32).
//   scatter(atomic f32, L2-resident) -> WMMA f32 GEMM (+bias, relu)
//   -> scatter -> WMMA f32 GEMM (+bias)
// Matrix ops use V_WMMA_F32_16X16X4_F32 (full f32 precision, 16x16 tile/wave).
// ---------------------------------------------------------------------------

typedef __attribute__((ext_vector_type(2))) float v2f;
typedef __attribute__((ext_vector_type(8))) float v8f;

#define N_NODES 100000
#define N_EDGES 1600000
#define IN_F    64
#define HID     128
#define N_CLS   64

// ---------------------------------------------------------------- zero fill
__global__ __launch_bounds__(256) void zero_f4(float4* __restrict__ p, long n4) {
    long i      = (long)blockIdx.x * blockDim.x + threadIdx.x;
    long stride = (long)gridDim.x * blockDim.x;
    float4 z = {0.f, 0.f, 0.f, 0.f};
    for (; i < n4; i += stride) p[i] = z;
}

// ------------------------------------------------------- edge scatter (SpMM)
// One (edge, feature) pair per loop iteration. Consecutive lanes cover
// consecutive features of the same edge -> coalesced gather loads and
// coalesced global_atomic_add_f32 that resolve in the 192MB L2.
template <int NF>
__global__ __launch_bounds__(256) void scatter_add(const float* __restrict__ feat,
                                                   const int*   __restrict__ src,
                                                   const int*   __restrict__ dst,
                                                   float*       __restrict__ agg,
                                                   long total) {
    long i      = (long)blockIdx.x * blockDim.x + threadIdx.x;
    long stride = (long)gridDim.x * blockDim.x;
    for (; i < total; i += stride) {
        int e = (int)(i / NF);   // NF is a power of two -> shift
        int f = (int)(i % NF);
        int s = src[e];
        int d = dst[e];
        float v = feat[(long)s * NF + f];
        atomicAdd(&agg[(long)d * NF + f], v);
    }
}

// ---------------------------------------------------- WMMA f32 dense GEMM
// out[M,NCOL] = relu?( A[M,K] @ W[K,NCOL] + bias )   with M % 16 == 0,
// NCOL % 16 == 0, K % 4 == 0. One wave per 16x16 output tile, K stepped by 4
// via V_WMMA_F32_16X16X4_F32.
//
// Per-lane layouts (CDNA5 ISA):
//   A 16x4 : lane l -> row m = l&15 ; VGPR0 = K(2*(l>>4)), VGPR1 = +1
//   B 4x16 : lane l -> col n = l&15 ; VGPR0 = K(2*(l>>4)), VGPR1 = +1
//   C/D    : VGPR j -> row j + 8*(l>>4), col l&15
template <int K, int NCOL, bool RELU>
__global__ __launch_bounds__(256) void gemm_wmma(const float* __restrict__ A,
                                                 const float* __restrict__ W,
                                                 const float* __restrict__ bias,
                                                 float*       __restrict__ out,
                                                 int nrows) {
    const int lane = threadIdx.x & 31;
    const int wid  = threadIdx.x >> 5;
    const int wavesPerBlock = blockDim.x >> 5;
    const int gw = blockIdx.x * wavesPerBlock + wid;

    constexpr int ntiles = NCOL / 16;
    const int rowTile = gw / ntiles;
    const int colTile = gw % ntiles;
    if (rowTile * 16 >= nrows) return;   // wave-uniform; EXEC stays all-ones

    const int m    = lane & 15;   // A row / B col / D col within tile
    const int kh   = lane >> 4;   // K half-select (0 or 1)
    const int row0 = rowTile * 16;
    const int n0   = colTile * 16;

    v8f c = {};
    const float* Arow = A + (long)(row0 + m) * K;

#pragma unroll
    for (int kk = 0; kk < K; kk += 4) {
        const int ka = kk + 2 * kh;            // even -> 8B-aligned float2 load
        v2f a = *(const v2f*)(Arow + ka);
        v2f b;
        b.x = W[(long)ka * NCOL + n0 + m];
        b.y = W[(long)(ka + 1) * NCOL + n0 + m];
        c = __builtin_amdgcn_wmma_f32_16x16x4_f32(
                /*neg_a=*/false, a, /*neg_b=*/false, b,
                /*c_mod=*/(short)0, c, /*reuse_a=*/false, /*reuse_b=*/false);
    }

    const float bv = bias[n0 + m];
#pragma unroll
    for (int j = 0; j < 8; ++j) {
        const int rr = row0 + j + 8 * kh;
        float v = c[j] + bv;
        if (RELU) v = fmaxf(v, 0.f);
        out[(long)rr * NCOL + n0 + m] = v;
    }
}

// ---------------------------------------------------------------------------
extern "C" void kernel_launch(void* const* d_in, const int* in_sizes, int n_in,
                              void* d_out, int out_size, void* d_ws, size_t ws_size,
                              hipStream_t stream) {
    (void)in_sizes; (void)n_in; (void)out_size; (void)ws_size;

    const float* x  = (const float*)d_in[0];
    const float* W1 = (const float*)d_in[1];
    const float* b1 = (const float*)d_in[2];
    const float* W2 = (const float*)d_in[3];
    const float* b2 = (const float*)d_in[4];
    const int*  src = (const int*)d_in[5];
    const int*  dst = (const int*)d_in[6];
    float*      out = (float*)d_out;

    // workspace: agg (reused both layers, sized for HID feats) then h
    float* agg = (float*)d_ws;                         // N_NODES * HID floats
    float* h   = agg + (size_t)N_NODES * HID;          // N_NODES * HID floats

    // ---- layer 1: agg1 = scatter(x), h = relu(agg1 @ W1 + b1) ----
    zero_f4<<<2048, 256, 0, stream>>>((float4*)agg, (long)N_NODES * IN_F / 4);
    scatter_add<IN_F><<<4096, 256, 0, stream>>>(x, src, dst, agg,
                                                (long)N_EDGES * IN_F);
    {
        const int waves  = (N_NODES / 16) * (HID / 16);   // 50000
        const int blocks = waves / 8;                     // 8 waves (256 thr)/block
        gemm_wmma<IN_F, HID, true><<<blocks, 256, 0, stream>>>(agg, W1, b1, h,
                                                               N_NODES);
    }

    // ---- layer 2: agg2 = scatter(h), out = agg2 @ W2 + b2 ----
    zero_f4<<<2048, 256, 0, stream>>>((float4*)agg, (long)N_NODES * HID / 4);
    scatter_add<HID><<<8192, 256, 0, stream>>>(h, src, dst, agg,
                                               (long)N_EDGES * HID);
    {
        const int waves  = (N_NODES / 16) * (N_CLS / 16); // 25000
        const int blocks = waves / 8;                     // 3125
        gemm_wmma<HID, N_CLS, false><<<blocks, 256, 0, stream>>>(agg, W2, b2,
                                                                 out, N_NODES);
    }
}